// BahdanauAttention_1047972020553
// MI455X (gfx1250) — compile-verified
//
#include <hip/hip_runtime.h>

typedef float v2f __attribute__((ext_vector_type(2)));
typedef float v8f __attribute__((ext_vector_type(8)));

#define H 32
#define SEQ 4096
#define BLOCK 512
#define WAVES (BLOCK / 32)            // 16 waves
#define POS_PER_WAVE (SEQ / WAVES)    // 256 positions per wave
#define TILES (POS_PER_WAVE / 16)     // 16 tiles of 16 positions

// Native CDNA5 V_TANH_F32 (trans unit, co-executes with WMMA). Fallback keeps
// the compile safe if the builtin name is absent in this toolchain.
#if defined(__has_builtin)
#if __has_builtin(__builtin_amdgcn_tanhf)
#define FAST_TANH(x) __builtin_amdgcn_tanhf(x)
#endif
#endif
#ifndef FAST_TANH
// branch-free fallback: tanh(x) = 1 - 2/(exp2(2*log2e*x)+1)
__device__ __forceinline__ float fast_tanh_fallback(float x) {
    float e = __expf(2.0f * x);
    return 1.0f - 2.0f / (e + 1.0f);
}
#define FAST_TANH(x) fast_tanh_fallback(x)
#endif

__global__ __launch_bounds__(BLOCK) void bahdanau_attn_kernel(
    const float* __restrict__ X,   // lstm_outputs [B,S,H]
    const float* __restrict__ FH,  // final_hidden [B,H]
    const float* __restrict__ Wq,  // [H,H]
    const float* __restrict__ bq,  // [H]
    const float* __restrict__ Wk,  // [H,H]
    const float* __restrict__ bk,  // [H]
    const float* __restrict__ Wv,  // [H,1]
    const float* __restrict__ bv,  // [1]
    float* __restrict__ out_ctx,   // [B,H]
    float* __restrict__ out_att)   // [B,S]
{
    __shared__ float s_scores[SEQ];              // 16 KB raw scores
    __shared__ float s_query[H];
    __shared__ float s_ctx[WAVES * 2 * 2 * 16];  // [wave][half][n][l16]
    __shared__ float s_M[WAVES * 2];             // running max per (wave,half)
    __shared__ float s_red[BLOCK];

    const int b    = blockIdx.x;
    const int tid  = threadIdx.x;
    const int wave = tid >> 5;
    const int lane = tid & 31;
    const int half = lane >> 4;
    const int l16  = lane & 15;

    // ---- query[d] = final_hidden[b] @ Wq + bq  (wave 0, lane = d) ----
    if (tid < H) {
        float fhv = FH[b * H + lane];
        float acc = bq[lane];
        #pragma unroll
        for (int h = 0; h < H; ++h) {
            float xh = __shfl(fhv, h, 32);
            acc = fmaf(xh, Wq[h * H + lane], acc);
        }
        s_query[lane] = acc;
    }

    // ---- Wk B-fragments, WMMA 16x16x4 B layout:
    //      element (K,N): lane = N%16 + 16*((K%4)>=2), vgpr = (K%4)&1
    v2f bf[8][2];
    #pragma unroll
    for (int j = 0; j < 8; ++j) {
        #pragma unroll
        for (int n = 0; n < 2; ++n) {
            const int k0  = 4 * j + 2 * half;
            const int col = n * 16 + l16;
            bf[j][n].x = Wk[(k0 + 0) * H + col];
            bf[j][n].y = Wk[(k0 + 1) * H + col];
        }
    }
    const float bk0 = bk[l16], bk1 = bk[l16 + 16];
    const float wv0 = Wv[l16], wv1 = Wv[l16 + 16];
    const float bvv = bv[0];

    __syncthreads();
    const float q0 = s_query[l16], q1 = s_query[l16 + 16];

    float ctx0 = 0.0f, ctx1 = 0.0f;          // online context, d = l16 (+16)
    float M = -3.402823466e38f;              // running max (uniform per half)

    const float* __restrict__ Xb = X + (size_t)b * SEQ * H;

    for (int it = 0; it < TILES; ++it) {
        const int s0 = wave * POS_PER_WAVE + it * 16;

        // A fragments: element (M,K): lane = M + 16*((K%4)>=2), vgpr=(K%4)&1
        const float* rowp = Xb + (size_t)(s0 + l16) * H + 2 * half;
        if (it + 1 < TILES)
            __builtin_prefetch((const void*)(rowp + 16 * H), 0, 3);
        v2f a[8];
        #pragma unroll
        for (int j = 0; j < 8; ++j)
            a[j] = __builtin_nontemporal_load((const v2f*)(rowp + 4 * j));

        v8f acc0 = {0.f,0.f,0.f,0.f,0.f,0.f,0.f,0.f};
        v8f acc1 = {0.f,0.f,0.f,0.f,0.f,0.f,0.f,0.f};
        #pragma unroll
        for (int j = 0; j < 8; ++j) {
            acc0 = __builtin_amdgcn_wmma_f32_16x16x4_f32(
                false, a[j], false, bf[j][0], (short)0, acc0, false, false);
            acc1 = __builtin_amdgcn_wmma_f32_16x16x4_f32(
                false, a[j], false, bf[j][1], (short)0, acc1, false, false);
        }

        // keys + bias, tanh, Wv partial dot. C/D layout: vgpr r, lane L ->
        // pos = r + 8*(L>>4), d = (L&15) + 16n
        float k0v[8], k1v[8], p[8];
        #pragma unroll
        for (int r = 0; r < 8; ++r) {
            k0v[r] = acc0[r] + bk0;
            k1v[r] = acc1[r] + bk1;
            p[r] = FAST_TANH(q0 + k0v[r]) * wv0 + FAST_TANH(q1 + k1v[r]) * wv1;
        }
        // reduce over d: across the 16 lanes of each half (masks stay in-half)
        #pragma unroll
        for (int off = 1; off < 16; off <<= 1) {
            #pragma unroll
            for (int r = 0; r < 8; ++r)
                p[r] += __shfl_xor(p[r], off, 32);
        }

        float sc[8];
        float m_new = M;
        #pragma unroll
        for (int r = 0; r < 8; ++r) {
            sc[r] = p[r] + bvv;                      // score of pos s0+8*half+r
            m_new = fmaxf(m_new, sc[r]);
            if (l16 == r) s_scores[s0 + 8 * half + r] = sc[r];
        }
        const float scale = __expf(M - m_new);
        ctx0 *= scale;
        ctx1 *= scale;
        #pragma unroll
        for (int r = 0; r < 8; ++r) {
            const float w = __expf(sc[r] - m_new);
            ctx0 = fmaf(w, k0v[r], ctx0);
            ctx1 = fmaf(w, k1v[r], ctx1);
        }
        M = m_new;
    }

    // stash per-(wave,half) unnormalized partials
    s_ctx[wave * 64 + half * 32 + 0  + l16] = ctx0;
    s_ctx[wave * 64 + half * 32 + 16 + l16] = ctx1;
    if (l16 == 0) s_M[wave * 2 + half] = M;
    __syncthreads();

    // ---- block-wide max over scores ----
    float mloc = -3.402823466e38f;
    #pragma unroll
    for (int i = 0; i < SEQ / BLOCK; ++i)
        mloc = fmaxf(mloc, s_scores[tid + i * BLOCK]);
    s_red[tid] = mloc;
    __syncthreads();
    for (int off = BLOCK / 2; off > 0; off >>= 1) {
        if (tid < off) s_red[tid] = fmaxf(s_red[tid], s_red[tid + off]);
        __syncthreads();
    }
    const float Mstar = s_red[0];
    __syncthreads();

    // ---- block-wide softmax denominator ----
    float dloc = 0.0f;
    #pragma unroll
    for (int i = 0; i < SEQ / BLOCK; ++i)
        dloc += __expf(s_scores[tid + i * BLOCK] - Mstar);
    s_red[tid] = dloc;
    __syncthreads();
    for (int off = BLOCK / 2; off > 0; off >>= 1) {
        if (tid < off) s_red[tid] += s_red[tid + off];
        __syncthreads();
    }
    const float Dinv = 1.0f / s_red[0];

    // ---- attention weights (coalesced, streamed once -> NT stores) ----
    for (int i = tid; i < SEQ; i += BLOCK) {
        float w = __expf(s_scores[i] - Mstar) * Dinv;
        __builtin_nontemporal_store(w, &out_att[(size_t)b * SEQ + i]);
    }

    // ---- context: combine partials with exp(M_wh - M*) correction ----
    if (tid < H) {
        const int n = tid >> 4, lo = tid & 15;
        float c = 0.0f;
        #pragma unroll
        for (int w = 0; w < WAVES; ++w) {
            #pragma unroll
            for (int hh = 0; hh < 2; ++hh) {
                c = fmaf(s_ctx[w * 64 + hh * 32 + n * 16 + lo],
                         __expf(s_M[w * 2 + hh] - Mstar), c);
            }
        }
        out_ctx[(size_t)b * H + tid] = c * Dinv;
    }
}

extern "C" void kernel_launch(void* const* d_in, const int* in_sizes, int n_in,
                              void* d_out, int out_size, void* d_ws, size_t ws_size,
                              hipStream_t stream) {
    const float* X  = (const float*)d_in[0];   // lstm_outputs [B,S,H]
    const float* FH = (const float*)d_in[1];   // final_hidden [B,H]
    const float* Wq = (const float*)d_in[2];
    const float* bq = (const float*)d_in[3];
    const float* Wk = (const float*)d_in[4];
    const float* bk = (const float*)d_in[5];
    const float* Wv = (const float*)d_in[6];
    const float* bv = (const float*)d_in[7];

    const int B = in_sizes[1] / H;             // final_hidden has B*H elements
    float* out_ctx = (float*)d_out;            // [B,H] first
    float* out_att = (float*)d_out + (size_t)B * H;  // then [B,S]

    bahdanau_attn_kernel<<<dim3(B), dim3(BLOCK), 0, stream>>>(
        X, FH, Wq, bq, Wk, bk, Wv, bv, out_ctx, out_att);
}